// ROI_19318762897865
// MI455X (gfx1250) — compile-verified
//
#include <hip/hip_runtime.h>

// ROI-Align + 2x2 maxpool for (B=4, N=512, C=256, Hf=Wf=50, OUT=14 -> 7x7).
// Store-bandwidth-bound (102.8 MB out @ 23.3 TB/s ~ 4.4us floor). Features
// (10 MB) are staged per-block into LDS with the CDNA5 Tensor Data Mover so
// the 4-point bilinear gathers hit LDS instead of issuing divergent L2 loads.

#define B_     4
#define N_     512
#define C_     256
#define HF     50
#define FSZ    (HF * HF)          // 2500 floats per channel map
#define ROI_PER_BLK 64
#define CH_PER_BLK  4

typedef unsigned int u32;
typedef u32 v4u __attribute__((ext_vector_type(4)));
typedef int  v8i __attribute__((ext_vector_type(8)));
typedef int  v4i __attribute__((ext_vector_type(4)));

__device__ __forceinline__ float roi_bilerp(const float* __restrict__ F,
                                            int rlo50, int rhi50, float rf,
                                            int clo, int chi, float cf) {
  float f00 = F[rlo50 + clo];
  float f01 = F[rlo50 + chi];
  float f10 = F[rhi50 + clo];
  float f11 = F[rhi50 + chi];
  float top = f00 * (1.0f - cf) + f01 * cf;
  float bot = f10 * (1.0f - cf) + f11 * cf;
  return top * (1.0f - rf) + bot * rf;
}

__global__ __launch_bounds__(256) void roi_align_tdm_kernel(
    const float* __restrict__ proposals,   // (B, N, 4)
    const float* __restrict__ features,    // (B, C, 50, 50)
    float* __restrict__ out)               // (B, N, C, 7, 7)
{
  __shared__ float feat[CH_PER_BLK * FSZ];                       // 40 KB
  __shared__ int   s_rlo[ROI_PER_BLK * 14], s_rhi[ROI_PER_BLK * 14];
  __shared__ int   s_clo[ROI_PER_BLK * 14], s_chi[ROI_PER_BLK * 14];
  __shared__ float s_rfr[ROI_PER_BLK * 14], s_cfr[ROI_PER_BLK * 14];

  const int tid = threadIdx.x;
  const int b   = blockIdx.z;
  const int c0  = blockIdx.y * CH_PER_BLK;
  const int n0  = blockIdx.x * ROI_PER_BLK;

  // ---- TDM: each of the 8 wave32s DMAs one 5 KB half-channel into LDS ----
  {
    u32 wave = (u32)tid >> 5;            // wave-uniform
    u32 chan = wave >> 1;                // 0..3
    u32 half = wave & 1u;                // 0..1
    u32 elem = chan * FSZ + half * (FSZ / 2);
    unsigned long long gaddr = (unsigned long long)features +
        (((unsigned long long)(b * C_ + c0) + chan) * FSZ + half * (FSZ / 2)) * 4ull;
    u32 laddr = (u32)(unsigned long long)(void*)feat + elem * 4u;

    // D# group 0: count=1, lds_addr, global_addr[56:0], type=2
    v4u g0;
    g0.x = 1u;
    g0.y = laddr;
    g0.z = (u32)gaddr;
    g0.w = (u32)((gaddr >> 32) & 0x1FFFFFFull) | (2u << 30);
    // D# group 1: data_size=2 (4B); tensor_dim0=1250; tensor_dim1=1;
    //             tile_dim0=1250; tile_dim1=1; dim0_stride=1250
    v8i g1;
    g1[0] = (int)(2u << 16);
    g1[1] = (int)((u32)(FSZ / 2) << 16);   // tensor_dim0 low 16 in [31:16]
    g1[2] = (int)(1u << 16);               // tensor_dim0 hi=0 | tensor_dim1 lo=1
    g1[3] = (int)((u32)(FSZ / 2) << 16);   // tensor_dim1 hi=0 | tile_dim0
    g1[4] = 1;                             // tile_dim1=1 | tile_dim2=0
    g1[5] = FSZ / 2;                       // tensor_dim0_stride low
    g1[6] = 0;
    g1[7] = 0;
    v4i gz = (v4i){0, 0, 0, 0};            // 2D tensor: groups 2/3 unused
#if __has_include(<hip/amd_detail/amd_gfx1250_TDM.h>)
    v8i gz8 = (v8i){0, 0, 0, 0, 0, 0, 0, 0};
    __builtin_amdgcn_tensor_load_to_lds(g0, g1, gz, gz, gz8, 0);
#else
    __builtin_amdgcn_tensor_load_to_lds(g0, g1, gz, gz, 0);
#endif
  }

  // ---- Per-ROI sampling tables (overlapped with the TDM transfer) ----
  if (tid < ROI_PER_BLK) {
    const float* P = proposals + ((size_t)(b * N_ + (n0 + tid))) * 4;
    const float fx = 50.0f / 800.0f;
    int x = (int)(P[0] * fx);
    int y = (int)(P[1] * fx);
    int w = (int)(P[2] * fx);
    int h = (int)(P[3] * fx);
    float sr = (float)(w - 1) / 13.0f;
    float sc = (float)(h - 1) / 13.0f;
    for (int i = 0; i < 14; ++i) {
      float cr = (float)i * sr;
      int   lo = (int)floorf(cr);
      float fr = cr - (float)lo;
      int   hi = min(lo + 1, w - 1);
      int rl = min(max(x + lo, 0), HF - 1);
      int rh = min(max(x + hi, 0), HF - 1);
      s_rlo[tid * 14 + i] = rl * HF;       // pre-scaled by row stride
      s_rhi[tid * 14 + i] = rh * HF;
      s_rfr[tid * 14 + i] = fr;

      float cc  = (float)i * sc;
      int   lo2 = (int)floorf(cc);
      float fc  = cc - (float)lo2;
      int   hi2 = min(lo2 + 1, h - 1);
      s_clo[tid * 14 + i] = min(max(y + lo2, 0), HF - 1);
      s_chi[tid * 14 + i] = min(max(y + hi2, 0), HF - 1);
      s_cfr[tid * 14 + i] = fc;
    }
  }

  __builtin_amdgcn_s_wait_tensorcnt(0);
  __syncthreads();

  // ---- One (roi, channel) pair per thread: 196 LDS bilerps -> 49 outputs --
  const int cl_ = tid & (CH_PER_BLK - 1);
  const int rl_ = tid / CH_PER_BLK;
  const float* __restrict__ F  = feat + cl_ * FSZ;
  const int*   __restrict__ RL = s_rlo + rl_ * 14;
  const int*   __restrict__ RH = s_rhi + rl_ * 14;
  const float* __restrict__ RF = s_rfr + rl_ * 14;
  const int*   __restrict__ CLo = s_clo + rl_ * 14;
  const int*   __restrict__ CHi = s_chi + rl_ * 14;
  const float* __restrict__ CF  = s_cfr + rl_ * 14;
  float* __restrict__ O = out +
      ((size_t)((b * N_ + (n0 + rl_)) * C_ + (c0 + cl_))) * 49;

  for (int ii = 0; ii < 7; ++ii) {
    int i0 = 2 * ii, i1 = i0 + 1;
    int   a0 = RL[i0], a1 = RH[i0];  float fr0 = RF[i0];
    int   b0 = RL[i1], b1 = RH[i1];  float fr1 = RF[i1];
    for (int jj = 0; jj < 7; ++jj) {
      int j0 = 2 * jj, j1 = j0 + 1;
      int   p0 = CLo[j0], q0 = CHi[j0];  float fc0 = CF[j0];
      int   p1 = CLo[j1], q1 = CHi[j1];  float fc1 = CF[j1];
      float v00 = roi_bilerp(F, a0, a1, fr0, p0, q0, fc0);
      float v01 = roi_bilerp(F, a0, a1, fr0, p1, q1, fc1);
      float v10 = roi_bilerp(F, b0, b1, fr1, p0, q0, fc0);
      float v11 = roi_bilerp(F, b0, b1, fr1, p1, q1, fc1);
      float m = fmaxf(fmaxf(v00, v01), fmaxf(v10, v11));
      __builtin_nontemporal_store(m, O + ii * 7 + jj);   // streaming output
    }
  }
}

extern "C" void kernel_launch(void* const* d_in, const int* in_sizes, int n_in,
                              void* d_out, int out_size, void* d_ws, size_t ws_size,
                              hipStream_t stream) {
  const float* proposals = (const float*)d_in[0];  // (4, 512, 4)
  const float* features  = (const float*)d_in[1];  // (4, 256, 50, 50)
  float* out = (float*)d_out;                      // (4, 512, 256, 7, 7)

  dim3 grid(N_ / ROI_PER_BLK, C_ / CH_PER_BLK, B_);  // 8 x 64 x 4
  roi_align_tdm_kernel<<<grid, 256, 0, stream>>>(proposals, features, out);
}